// PositionWiseMixtureOfExpertsNetwork_50955492000006
// MI455X (gfx1250) — compile-verified
//
#include <hip/hip_runtime.h>

typedef float v2f __attribute__((ext_vector_type(2)));
typedef float v8f __attribute__((ext_vector_type(8)));

#define D_DIM 1024
#define F_DIM 4096
#define E_NUM 8
#define N_TOK 2048            // B*S = 2*1024
#define MAX_TILES 136         // N_TOK/16 + E_NUM (worst-case padded tiles)

// One 16x16 fp32 output tile per wave32: C += A(16xK) * B(Kx16) via V_WMMA_F32_16X16X4_F32.
// arow: per-lane pointer to A row (row = lane&15), contiguous K.
// bcol: per-lane pointer to B column (col = lane&15), stride ldb between K rows.
// koff: (lane>>4)*2  (upper half-wave holds K+2/K+3 per the f32 A/B layout)
__device__ __forceinline__ v8f wave_gemm_f32(const float* __restrict__ arow,
                                             const float* __restrict__ bcol,
                                             int ldb, int K, int koff) {
  v8f acc = {0.f, 0.f, 0.f, 0.f, 0.f, 0.f, 0.f, 0.f};
#pragma unroll 8
  for (int k = 0; k < K; k += 4) {
    v2f a = *(const v2f*)(arow + k + koff);           // global_load_b64
    v2f b;
    b.x = bcol[(size_t)(k + koff) * ldb];
    b.y = bcol[(size_t)(k + koff + 1) * ldb];
    acc = __builtin_amdgcn_wmma_f32_16x16x4_f32(
        /*neg_a=*/false, a, /*neg_b=*/false, b,
        /*c_mod=*/(short)0, acc, /*reuse_a=*/false, /*reuse_b=*/false);
  }
  return acc;
}

// ---- Kernel 1: router layer 1: H = relu(X @ W1 + b1), X[N,D], W1[D,D] ----
__global__ __launch_bounds__(256) void router_gemm1(const float* __restrict__ X,
                                                    const float* __restrict__ W1,
                                                    const float* __restrict__ b1,
                                                    float* __restrict__ H) {
  const int lane = threadIdx.x & 31;
  const int wave = threadIdx.x >> 5;
  const int tileN = blockIdx.x * 8 + wave;   // D/16 = 64 tiles -> gridDim.x = 8
  const int tileM = blockIdx.y;              // N/16 = 128
  const int row   = lane & 15;
  const int koff  = (lane >> 4) << 1;
  const int col   = tileN * 16 + (lane & 15);

  const float* arow = X + (size_t)(tileM * 16 + row) * D_DIM;
  const float* bcol = W1 + col;
  v8f acc = wave_gemm_f32(arow, bcol, D_DIM, D_DIM, koff);

  const float bias = b1[col];
  const int rbase = tileM * 16 + ((lane >> 4) << 3);
#pragma unroll
  for (int r = 0; r < 8; ++r) {
    float v = acc[r] + bias;
    H[(size_t)(rbase + r) * D_DIM + col] = v > 0.f ? v : 0.f;
  }
}

// ---- Kernel 2: router layer 2 + argmax (softmax is monotone) ----
__global__ __launch_bounds__(128) void router2_argmax(const float* __restrict__ H,
                                                      const float* __restrict__ W2,
                                                      const float* __restrict__ b2,
                                                      int* __restrict__ eidx) {
  const int token = blockIdx.x;
  const int tid = threadIdx.x;
  __shared__ float s[E_NUM * 128];
  __shared__ float logits[E_NUM];

  float acc[E_NUM];
#pragma unroll
  for (int e = 0; e < E_NUM; ++e) acc[e] = 0.f;

  const float* hrow = H + (size_t)token * D_DIM;
  for (int d = tid; d < D_DIM; d += 128) {
    const float hv = hrow[d];
    const float* w = W2 + (size_t)d * E_NUM;
#pragma unroll
    for (int e = 0; e < E_NUM; ++e) acc[e] += hv * w[e];
  }
#pragma unroll
  for (int e = 0; e < E_NUM; ++e) s[e * 128 + tid] = acc[e];
  __syncthreads();
  if (tid < E_NUM) {
    float sum = 0.f;
    for (int i = 0; i < 128; ++i) sum += s[tid * 128 + i];
    logits[tid] = sum + b2[tid];
  }
  __syncthreads();
  if (tid == 0) {
    int best = 0;
    float bv = logits[0];
    for (int e = 1; e < E_NUM; ++e) {      // strict > keeps first-index tie (jnp.argmax)
      if (logits[e] > bv) { bv = logits[e]; best = e; }
    }
    eidx[token] = best;
  }
}

// ---- Token grouping: counts / scan / scatter ----
__global__ void moe_zero(int* __restrict__ counts, int* __restrict__ cursors) {
  const int t = threadIdx.x;
  if (t < E_NUM) { counts[t] = 0; cursors[t] = 0; }
}

__global__ void moe_count(const int* __restrict__ eidx, int* __restrict__ counts) {
  const int t = blockIdx.x * blockDim.x + threadIdx.x;
  if (t < N_TOK) atomicAdd(&counts[eidx[t]], 1);
}

__global__ void moe_scan(const int* __restrict__ counts, int* __restrict__ offsets,
                         int* __restrict__ tileOff) {
  if (threadIdx.x == 0) {
    int o = 0, to = 0;
    for (int e = 0; e < E_NUM; ++e) {
      offsets[e] = o;
      tileOff[e] = to;
      o  += counts[e];
      to += (counts[e] + 15) >> 4;
    }
    offsets[E_NUM] = o;
    tileOff[E_NUM] = to;
  }
}

__global__ void moe_scatter(const int* __restrict__ eidx, const int* __restrict__ offsets,
                            int* __restrict__ cursors, int* __restrict__ perm) {
  const int t = blockIdx.x * blockDim.x + threadIdx.x;
  if (t < N_TOK) {
    const int e = eidx[t];
    const int p = atomicAdd(&cursors[e], 1);
    perm[offsets[e] + p] = t;
  }
}

// ---- Kernel 7: expert layer 1: HID[g,:] = relu(X[perm[g],:] @ eW1[e] + eB1[e]) ----
__global__ __launch_bounds__(256) void expert_ffn1(const float* __restrict__ X,
                                                   const float* __restrict__ eW1,
                                                   const float* __restrict__ eB1,
                                                   const int* __restrict__ perm,
                                                   const int* __restrict__ offsets,
                                                   const int* __restrict__ tileOff,
                                                   float* __restrict__ HID) {
  const int flat = blockIdx.y;
  if (flat >= tileOff[E_NUM]) return;          // wave-uniform
  int e = 0;
  while (flat >= tileOff[e + 1]) e++;          // <= 7 iterations, wave-uniform
  const int base = offsets[e];
  const int lim  = offsets[e + 1];
  const int lt   = flat - tileOff[e];

  const int lane = threadIdx.x & 31;
  const int wave = threadIdx.x >> 5;
  const int tileF = blockIdx.x * 8 + wave;     // F/16 = 256 tiles -> gridDim.x = 32
  const int row  = lane & 15;
  const int koff = (lane >> 4) << 1;
  const int col  = tileF * 16 + (lane & 15);

  int gpos = base + lt * 16 + row;
  if (gpos >= lim) gpos = lim - 1;             // clamp padded rows to a valid token
  const int token = perm[gpos];

  const float* arow = X + (size_t)token * D_DIM;
  const float* bcol = eW1 + (size_t)e * D_DIM * F_DIM + col;   // eW1[e][k][f]
  v8f acc = wave_gemm_f32(arow, bcol, F_DIM, D_DIM, koff);

  const float bias = eB1[e * F_DIM + col];
  const int rbase = base + lt * 16 + ((lane >> 4) << 3);
#pragma unroll
  for (int r = 0; r < 8; ++r) {
    const int g = rbase + r;
    if (g < lim) {
      float v = acc[r] + bias;
      HID[(size_t)g * F_DIM + col] = v > 0.f ? v : 0.f;
    }
  }
}

// ---- Kernel 8: expert layer 2 + scatter: OUT[perm[g],:] = HID[g,:] @ eW2[e] + eB2[e] ----
__global__ __launch_bounds__(256) void expert_ffn2(const float* __restrict__ HID,
                                                   const float* __restrict__ eW2,
                                                   const float* __restrict__ eB2,
                                                   const int* __restrict__ perm,
                                                   const int* __restrict__ offsets,
                                                   const int* __restrict__ tileOff,
                                                   float* __restrict__ OUT) {
  const int flat = blockIdx.y;
  if (flat >= tileOff[E_NUM]) return;
  int e = 0;
  while (flat >= tileOff[e + 1]) e++;
  const int base = offsets[e];
  const int lim  = offsets[e + 1];
  const int lt   = flat - tileOff[e];

  const int lane = threadIdx.x & 31;
  const int wave = threadIdx.x >> 5;
  const int tileD = blockIdx.x * 8 + wave;     // D/16 = 64 tiles -> gridDim.x = 8
  const int row  = lane & 15;
  const int koff = (lane >> 4) << 1;
  const int col  = tileD * 16 + (lane & 15);

  int gpos = base + lt * 16 + row;
  if (gpos >= lim) gpos = lim - 1;

  const float* arow = HID + (size_t)gpos * F_DIM;
  const float* bcol = eW2 + (size_t)e * F_DIM * D_DIM + col;   // eW2[e][f][d]
  v8f acc = wave_gemm_f32(arow, bcol, D_DIM, F_DIM, koff);

  const float bias = eB2[e * D_DIM + col];
  const int rbase = base + lt * 16 + ((lane >> 4) << 3);
#pragma unroll
  for (int r = 0; r < 8; ++r) {
    const int g = rbase + r;
    if (g < lim) {
      const int token = perm[g];
      OUT[(size_t)token * D_DIM + col] = acc[r] + bias;
    }
  }
}

extern "C" void kernel_launch(void* const* d_in, const int* in_sizes, int n_in,
                              void* d_out, int out_size, void* d_ws, size_t ws_size,
                              hipStream_t stream) {
  (void)in_sizes; (void)n_in; (void)out_size; (void)ws_size;
  const float* X   = (const float*)d_in[0];  // [B,S,D] = [N_TOK, D]
  const float* rW1 = (const float*)d_in[1];  // [D,D]
  const float* rb1 = (const float*)d_in[2];  // [D]
  const float* rW2 = (const float*)d_in[3];  // [D,E]
  const float* rb2 = (const float*)d_in[4];  // [E]
  const float* eW1 = (const float*)d_in[5];  // [E,D,F]
  const float* eB1 = (const float*)d_in[6];  // [E,F]
  const float* eW2 = (const float*)d_in[7];  // [E,F,D]
  const float* eB2 = (const float*)d_in[8];  // [E,D]
  float* OUT = (float*)d_out;

  // Workspace layout: H (8 MB) | HID (32 MB) | int metadata
  float* H   = (float*)d_ws;
  float* HID = H + (size_t)N_TOK * D_DIM;
  int* meta    = (int*)(HID + (size_t)N_TOK * F_DIM);
  int* eidx    = meta;                 // N_TOK
  int* perm    = meta + N_TOK;         // N_TOK
  int* counts  = meta + 2 * N_TOK;     // E
  int* cursors = counts + E_NUM;       // E
  int* offsets = cursors + E_NUM;      // E+1
  int* tileOff = offsets + (E_NUM + 1);// E+1

  router_gemm1<<<dim3(8, 128), 256, 0, stream>>>(X, rW1, rb1, H);
  router2_argmax<<<N_TOK, 128, 0, stream>>>(H, rW2, rb2, eidx);
  moe_zero<<<1, 32, 0, stream>>>(counts, cursors);
  moe_count<<<N_TOK / 256, 256, 0, stream>>>(eidx, counts);
  moe_scan<<<1, 32, 0, stream>>>(counts, offsets, tileOff);
  moe_scatter<<<N_TOK / 256, 256, 0, stream>>>(eidx, offsets, cursors, perm);
  expert_ffn1<<<dim3(32, MAX_TILES), 256, 0, stream>>>(X, eW1, eB1, perm, offsets, tileOff, HID);
  expert_ffn2<<<dim3(8, MAX_TILES), 256, 0, stream>>>(HID, eW2, eB2, perm, offsets, tileOff, OUT);
}